// ZCANormSVDPI_6992206758142
// MI455X (gfx1250) — compile-verified
//
#include <hip/hip_runtime.h>
#include <math.h>

// ---------------- problem constants ----------------
#define C_DIM   2048
#define G_DIM   4
#define L_DIM   512
#define N_DIM   16384
#define EPS_F   1e-4f
#define NS_ITERS 8

typedef __attribute__((ext_vector_type(2))) float v2f;
typedef __attribute__((ext_vector_type(4))) float v4f;
typedef __attribute__((ext_vector_type(8))) float v8f;

// LDS tile stride (64 + 4 pad to stagger banks)
#define LDSS 68

// ---------------- workspace layout (in floats) ----------------
#define GLL     (G_DIM * L_DIM * L_DIM)        // 1,048,576 floats / matrix set
#define WS_MEAN 0                              // C_DIM floats
#define WS_TR   (C_DIM)                        // G_DIM floats
#define WS_COV  4096                           // GLL
#define WS_Y0   (WS_COV + GLL)
#define WS_Y1   (WS_Y0 + GLL)
#define WS_Z0   (WS_Y1 + GLL)
#define WS_Z1   (WS_Z0 + GLL)
#define WS_T    (WS_Z1 + GLL)
// total: 4096 + 6*GLL floats ~= 25.2 MB

// ---------------- small helpers ----------------
__device__ __forceinline__ v4f vsub(v4f a, float m) {
  v4f r; r.x = a.x - m; r.y = a.y - m; r.z = a.z - m; r.w = a.w - m; return r;
}

// load 8 v4f rows (row step = rowstride floats) into registers
__device__ __forceinline__ void load8(const float* __restrict__ p, size_t rowstride,
                                      v4f r[8]) {
  #pragma unroll
  for (int i = 0; i < 8; ++i) r[i] = *(const v4f*)(p + (size_t)i * rowstride);
}
// store 8 register rows to LDS (row step = 8*LDSS)
__device__ __forceinline__ void store8(float* s, const v4f r[8]) {
  #pragma unroll
  for (int i = 0; i < 8; ++i) *(v4f*)(s + i * 8 * LDSS) = r[i];
}
__device__ __forceinline__ void store8_sub(float* s, const v4f r[8], const float m[8]) {
  #pragma unroll
  for (int i = 0; i < 8; ++i) *(v4f*)(s + i * 8 * LDSS) = vsub(r[i], m[i]);
}

// ==================================================================
// Wave-level inner loops over one 64-wide K chunk, 2x2 register tile.
//   _nn : both panels n/m-major rows with contiguous k  (cov)
//   _kn : A m-major rows, B k-major rows                (gemm, apply)
// ==================================================================
__device__ __forceinline__ void wmma_chunk_nn(const float* __restrict__ As,
                                              const float* __restrict__ Bs,
                                              int am0, int am1, int bn0, int bn1,
                                              int kb,
                                              v8f& c00, v8f& c01, v8f& c10, v8f& c11) {
  #pragma unroll
  for (int k4 = 0; k4 < 64; k4 += 4) {
    v2f a0, a1, b0, b1;
    a0.x = As[am0 * LDSS + k4 + kb + 0];
    a0.y = As[am0 * LDSS + k4 + kb + 1];
    a1.x = As[am1 * LDSS + k4 + kb + 0];
    a1.y = As[am1 * LDSS + k4 + kb + 1];
    b0.x = Bs[bn0 * LDSS + k4 + kb + 0];
    b0.y = Bs[bn0 * LDSS + k4 + kb + 1];
    b1.x = Bs[bn1 * LDSS + k4 + kb + 0];
    b1.y = Bs[bn1 * LDSS + k4 + kb + 1];
    c00 = __builtin_amdgcn_wmma_f32_16x16x4_f32(false, a0, false, b0, (short)0, c00, false, false);
    c01 = __builtin_amdgcn_wmma_f32_16x16x4_f32(false, a0, false, b1, (short)0, c01, false, false);
    c10 = __builtin_amdgcn_wmma_f32_16x16x4_f32(false, a1, false, b0, (short)0, c10, false, false);
    c11 = __builtin_amdgcn_wmma_f32_16x16x4_f32(false, a1, false, b1, (short)0, c11, false, false);
  }
}

__device__ __forceinline__ void wmma_chunk_kn(const float* __restrict__ As,
                                              const float* __restrict__ Bs,
                                              int am0, int am1, int bn0, int bn1,
                                              int kb,
                                              v8f& c00, v8f& c01, v8f& c10, v8f& c11) {
  #pragma unroll
  for (int k4 = 0; k4 < 64; k4 += 4) {
    v2f a0, a1, b0, b1;
    a0.x = As[am0 * LDSS + k4 + kb + 0];
    a0.y = As[am0 * LDSS + k4 + kb + 1];
    a1.x = As[am1 * LDSS + k4 + kb + 0];
    a1.y = As[am1 * LDSS + k4 + kb + 1];
    b0.x = Bs[(k4 + kb + 0) * LDSS + bn0];
    b0.y = Bs[(k4 + kb + 1) * LDSS + bn0];
    b1.x = Bs[(k4 + kb + 0) * LDSS + bn1];
    b1.y = Bs[(k4 + kb + 1) * LDSS + bn1];
    c00 = __builtin_amdgcn_wmma_f32_16x16x4_f32(false, a0, false, b0, (short)0, c00, false, false);
    c01 = __builtin_amdgcn_wmma_f32_16x16x4_f32(false, a0, false, b1, (short)0, c01, false, false);
    c10 = __builtin_amdgcn_wmma_f32_16x16x4_f32(false, a1, false, b0, (short)0, c10, false, false);
    c11 = __builtin_amdgcn_wmma_f32_16x16x4_f32(false, a1, false, b1, (short)0, c11, false, false);
  }
}

// wave/lane -> fragment coordinates (block = 128 thr = 4 waves, 64x64 tile,
// each wave owns a 32x32 quadrant)
struct WaveCoord {
  int am0, am1, bn0, bn1, kb;   // LDS fragment indices
  int mb0, mb1, nc0, nc1;       // epilogue D coords (local to 64x64 tile)
};
__device__ __forceinline__ WaveCoord wave_coord(int tid) {
  WaveCoord w;
  const int lane = tid & 31;
  const int wave = tid >> 5;
  const int wm = (wave >> 1) << 5;   // 0 or 32
  const int wn = (wave & 1) << 5;    // 0 or 32
  w.am0 = wm + (lane & 15);
  w.am1 = w.am0 + 16;
  w.bn0 = wn + (lane & 15);
  w.bn1 = w.bn0 + 16;
  w.kb  = (lane >> 4) << 1;          // lanes>=16 take k+2,k+3
  w.mb0 = wm + 8 * (lane >> 4);      // D: vgpr j -> M = j (+8 for hi lanes)
  w.mb1 = w.mb0 + 16;
  w.nc0 = wn + (lane & 15);
  w.nc1 = w.nc0 + 16;
  return w;
}

// ==================================================================
// Kernel 1: per-channel mean over N (one block per channel)
// ==================================================================
__global__ void __launch_bounds__(256) mean_kernel(const float* __restrict__ x,
                                                   float* __restrict__ ws) {
  const int c = blockIdx.x;
  const v4f* row = (const v4f*)(x + (size_t)c * N_DIM);
  float s = 0.0f;
  #pragma unroll
  for (int j = 0; j < 16; ++j) {
    v4f v = row[threadIdx.x + 256 * j];
    s += v.x + v.y + v.z + v.w;
  }
  #pragma unroll
  for (int off = 16; off > 0; off >>= 1) s += __shfl_down(s, off);
  __shared__ float red[8];
  if ((threadIdx.x & 31) == 0) red[threadIdx.x >> 5] = s;
  __syncthreads();
  if (threadIdx.x == 0) {
    float t = 0.0f;
    #pragma unroll
    for (int i = 0; i < 8; ++i) t += red[i];
    ws[WS_MEAN + c] = t * (1.0f / (float)N_DIM);
  }
}

// ==================================================================
// Kernel 2: per-group covariance cov = xc*xc^T / N + eps*I
// block = 128 thr (4 waves), 64x64 tile; grid = G_DIM * 64
// Both LDS panels n/m-major: all fragment reads are contiguous k-pairs.
// Software-pipelined: next chunk prefetched to regs during compute.
// ==================================================================
__global__ void __launch_bounds__(128) cov_kernel(const float* __restrict__ x,
                                                  float* __restrict__ ws) {
  __shared__ float As[64 * LDSS];
  __shared__ float Bs[64 * LDSS];
  const float* mean = ws + WS_MEAN;

  const int blk = blockIdx.x;
  const int g   = blk >> 6;
  const int t   = blk & 63;
  const int tileM = (t >> 3) << 6;
  const int tileN = (t & 7) << 6;
  const int gL  = g * L_DIM;

  const int tid = threadIdx.x;
  const WaveCoord wc = wave_coord(tid);
  const int lr = tid >> 4;           // loader row 0..7
  const int lc = (tid & 15) << 2;    // loader col 0..60 step 4

  // loop-invariant row means
  float ma[8], mb[8];
  #pragma unroll
  for (int i = 0; i < 8; ++i) {
    ma[i] = mean[gL + tileM + lr + 8 * i];
    mb[i] = mean[gL + tileN + lr + 8 * i];
  }

  const float* pA = x + (size_t)(gL + tileM + lr) * N_DIM + lc;
  const float* pB = x + (size_t)(gL + tileN + lr) * N_DIM + lc;
  float* sA = &As[lr * LDSS + lc];
  float* sB = &Bs[lr * LDSS + lc];

  v8f c00 = {}, c01 = {}, c10 = {}, c11 = {};
  v4f ra[8], rb[8];
  load8(pA, (size_t)8 * N_DIM, ra);
  load8(pB, (size_t)8 * N_DIM, rb);

  for (int kk = 0; kk < N_DIM; kk += 64) {
    __syncthreads();
    store8_sub(sA, ra, ma);
    store8_sub(sB, rb, mb);
    __syncthreads();
    if (kk + 64 < N_DIM) {           // prefetch next chunk; overlaps compute
      load8(pA + kk + 64, (size_t)8 * N_DIM, ra);
      load8(pB + kk + 64, (size_t)8 * N_DIM, rb);
    }
    wmma_chunk_nn(As, Bs, wc.am0, wc.am1, wc.bn0, wc.bn1, wc.kb, c00, c01, c10, c11);
  }

  const float invN = 1.0f / (float)N_DIM;
  float* covg = ws + WS_COV + (size_t)g * L_DIM * L_DIM;
  #pragma unroll
  for (int j = 0; j < 8; ++j) {
    const int m0 = tileM + wc.mb0 + j, m1 = tileM + wc.mb1 + j;
    const int n0 = tileN + wc.nc0,     n1 = tileN + wc.nc1;
    float v00 = c00[j] * invN; if (m0 == n0) v00 += EPS_F;
    float v01 = c01[j] * invN; if (m0 == n1) v01 += EPS_F;
    float v10 = c10[j] * invN; if (m1 == n0) v10 += EPS_F;
    float v11 = c11[j] * invN; if (m1 == n1) v11 += EPS_F;
    covg[(size_t)m0 * L_DIM + n0] = v00;
    covg[(size_t)m0 * L_DIM + n1] = v01;
    covg[(size_t)m1 * L_DIM + n0] = v10;
    covg[(size_t)m1 * L_DIM + n1] = v11;
  }
}

// ==================================================================
// Kernel 3: trace of cov per group (one block per group)
// ==================================================================
__global__ void __launch_bounds__(256) trace_kernel(float* __restrict__ ws) {
  const int g = blockIdx.x;
  const float* covg = ws + WS_COV + (size_t)g * L_DIM * L_DIM;
  float s = 0.0f;
  for (int i = threadIdx.x; i < L_DIM; i += 256)
    s += covg[(size_t)i * L_DIM + i];
  #pragma unroll
  for (int off = 16; off > 0; off >>= 1) s += __shfl_down(s, off);
  __shared__ float red[8];
  if ((threadIdx.x & 31) == 0) red[threadIdx.x >> 5] = s;
  __syncthreads();
  if (threadIdx.x == 0) {
    float t = 0.0f;
    #pragma unroll
    for (int i = 0; i < 8; ++i) t += red[i];
    ws[WS_TR + g] = t;
  }
}

// ==================================================================
// Kernel 4: Newton-Schulz init:  Y0 = cov/tr,  Z0 = I
// ==================================================================
__global__ void __launch_bounds__(256) ns_init_kernel(float* __restrict__ ws) {
  const size_t idx = (size_t)blockIdx.x * 256 + threadIdx.x;
  const int g  = (int)(idx / ((size_t)L_DIM * L_DIM));
  const int ij = (int)(idx % ((size_t)L_DIM * L_DIM));
  const int i  = ij / L_DIM;
  const int j  = ij % L_DIM;
  const float invtr = 1.0f / ws[WS_TR + g];
  ws[WS_Y0 + idx] = ws[WS_COV + idx] * invtr;
  ws[WS_Z0 + idx] = (i == j) ? 1.0f : 0.0f;
}

// ==================================================================
// Kernel 5: batched 512^3 GEMM over G groups (WMMA f32)
//   MODE 0 : D = 3I - A@B      (NS "T" step)
//   MODE 1 : D = 0.5 * A@B     (NS "Y"/"Z" steps)
// block = 128 thr, 64x64 tile; grid = G_DIM*64; pipelined staging
// ==================================================================
template <int MODE>
__global__ void __launch_bounds__(128) gemm512_kernel(const float* __restrict__ Am,
                                                      const float* __restrict__ Bm,
                                                      float* __restrict__ Dm) {
  __shared__ float As[64 * LDSS];
  __shared__ float Bs[64 * LDSS];

  const int blk = blockIdx.x;
  const int g   = blk >> 6;
  const int t   = blk & 63;
  const int tileM = (t >> 3) << 6;
  const int tileN = (t & 7) << 6;

  const float* A = Am + (size_t)g * L_DIM * L_DIM;
  const float* B = Bm + (size_t)g * L_DIM * L_DIM;
  float*       D = Dm + (size_t)g * L_DIM * L_DIM;

  const int tid = threadIdx.x;
  const WaveCoord wc = wave_coord(tid);
  const int lr = tid >> 4;
  const int lc = (tid & 15) << 2;

  const float* pA = A + (size_t)(tileM + lr) * L_DIM + lc;   // advance by +kk cols
  const float* pB = B + (size_t)lr * L_DIM + tileN + lc;     // advance by +kk rows
  float* sA = &As[lr * LDSS + lc];
  float* sB = &Bs[lr * LDSS + lc];

  v8f c00 = {}, c01 = {}, c10 = {}, c11 = {};
  v4f ra[8], rb[8];
  load8(pA, (size_t)8 * L_DIM, ra);
  load8(pB, (size_t)8 * L_DIM, rb);

  for (int kk = 0; kk < L_DIM; kk += 64) {
    __syncthreads();
    store8(sA, ra);
    store8(sB, rb);
    __syncthreads();
    if (kk + 64 < L_DIM) {
      load8(pA + kk + 64, (size_t)8 * L_DIM, ra);
      load8(pB + (size_t)(kk + 64) * L_DIM, (size_t)8 * L_DIM, rb);
    }
    wmma_chunk_kn(As, Bs, wc.am0, wc.am1, wc.bn0, wc.bn1, wc.kb, c00, c01, c10, c11);
  }

  #pragma unroll
  for (int j = 0; j < 8; ++j) {
    const int m0 = tileM + wc.mb0 + j, m1 = tileM + wc.mb1 + j;
    const int n0 = tileN + wc.nc0,     n1 = tileN + wc.nc1;
    float v00, v01, v10, v11;
    if (MODE == 0) {
      v00 = ((m0 == n0) ? 3.0f : 0.0f) - c00[j];
      v01 = ((m0 == n1) ? 3.0f : 0.0f) - c01[j];
      v10 = ((m1 == n0) ? 3.0f : 0.0f) - c10[j];
      v11 = ((m1 == n1) ? 3.0f : 0.0f) - c11[j];
    } else {
      v00 = 0.5f * c00[j]; v01 = 0.5f * c01[j];
      v10 = 0.5f * c10[j]; v11 = 0.5f * c11[j];
    }
    D[(size_t)m0 * L_DIM + n0] = v00;
    D[(size_t)m0 * L_DIM + n1] = v01;
    D[(size_t)m1 * L_DIM + n0] = v10;
    D[(size_t)m1 * L_DIM + n1] = v11;
  }
}

// ==================================================================
// Kernel 6: apply whitening:
//   out[c,n] = (Zf[g] @ xc[g])[c',n] * rsqrt(tr[g]) * w[c] + b[c]
// block = 128 thr, 64(chan) x 64(sample) tile; grid = G_DIM*8*256
// Pipelined staging; chunk-dependent means prefetched with B regs.
// ==================================================================
__global__ void __launch_bounds__(128) apply_kernel(const float* __restrict__ x,
                                                    const float* __restrict__ weight,
                                                    const float* __restrict__ bias,
                                                    const float* __restrict__ ws,
                                                    float* __restrict__ out) {
  __shared__ float As[64 * LDSS];
  __shared__ float Bs[64 * LDSS];

  const int blk = blockIdx.x;
  const int g   = blk >> 11;            // 2048 tiles per group
  const int rem = blk & 2047;
  const int tileM = (rem >> 8) << 6;    // 8 M-tiles
  const int tileN = (rem & 255) << 6;   // 256 N-tiles
  const int gL = g * L_DIM;

  const float* Z    = ws + WS_Z0 + (size_t)g * L_DIM * L_DIM;  // even #iters -> Z0
  const float* mean = ws + WS_MEAN;
  const float  scale = rsqrtf(ws[WS_TR + g]);

  const int tid = threadIdx.x;
  const WaveCoord wc = wave_coord(tid);
  const int lr = tid >> 4;
  const int lc = (tid & 15) << 2;

  const float* pA = Z + (size_t)(tileM + lr) * L_DIM + lc;     // advance by +kk cols
  const float* pB = x + (size_t)(gL + lr) * N_DIM + tileN + lc; // advance by +kk rows
  float* sA = &As[lr * LDSS + lc];
  float* sB = &Bs[lr * LDSS + lc];

  v8f c00 = {}, c01 = {}, c10 = {}, c11 = {};
  v4f ra[8], rb[8];
  float mb[8];
  load8(pA, (size_t)8 * L_DIM, ra);
  load8(pB, (size_t)8 * N_DIM, rb);
  #pragma unroll
  for (int i = 0; i < 8; ++i) mb[i] = mean[gL + lr + 8 * i];

  for (int kk = 0; kk < L_DIM; kk += 64) {
    __syncthreads();
    store8(sA, ra);
    store8_sub(sB, rb, mb);
    __syncthreads();
    if (kk + 64 < L_DIM) {
      load8(pA + kk + 64, (size_t)8 * L_DIM, ra);
      load8(pB + (size_t)(kk + 64) * N_DIM, (size_t)8 * N_DIM, rb);
      #pragma unroll
      for (int i = 0; i < 8; ++i) mb[i] = mean[gL + kk + 64 + lr + 8 * i];
    }
    wmma_chunk_kn(As, Bs, wc.am0, wc.am1, wc.bn0, wc.bn1, wc.kb, c00, c01, c10, c11);
  }

  #pragma unroll
  for (int j = 0; j < 8; ++j) {
    const int m0 = tileM + wc.mb0 + j, m1 = tileM + wc.mb1 + j;
    const int n0 = tileN + wc.nc0,     n1 = tileN + wc.nc1;
    const int ch0 = gL + m0, ch1 = gL + m1;
    const float w0 = weight[ch0] * scale, b0 = bias[ch0];
    const float w1 = weight[ch1] * scale, b1 = bias[ch1];
    out[(size_t)ch0 * N_DIM + n0] = c00[j] * w0 + b0;
    out[(size_t)ch0 * N_DIM + n1] = c01[j] * w0 + b0;
    out[(size_t)ch1 * N_DIM + n0] = c10[j] * w1 + b1;
    out[(size_t)ch1 * N_DIM + n1] = c11[j] * w1 + b1;
  }
}

// ==================================================================
// host-side launch sequence (graph-capture safe: fixed launches, no alloc)
// ==================================================================
extern "C" void kernel_launch(void* const* d_in, const int* in_sizes, int n_in,
                              void* d_out, int out_size, void* d_ws, size_t ws_size,
                              hipStream_t stream) {
  (void)in_sizes; (void)n_in; (void)out_size; (void)ws_size;

  const float* x      = (const float*)d_in[0];   // (C, N) f32
  const float* weight = (const float*)d_in[1];   // (C, 1) f32
  const float* bias   = (const float*)d_in[2];   // (C, 1) f32
  float* out = (float*)d_out;
  float* ws  = (float*)d_ws;

  // 1. per-channel means
  mean_kernel<<<C_DIM, 256, 0, stream>>>(x, ws);

  // 2. grouped covariance (WMMA f32 GEMM, K = N)
  cov_kernel<<<G_DIM * 64, 128, 0, stream>>>(x, ws);

  // 3. trace normalization + NS init
  trace_kernel<<<G_DIM, 256, 0, stream>>>(ws);
  ns_init_kernel<<<GLL / 256, 256, 0, stream>>>(ws);

  // 4. coupled Newton-Schulz iterations for cov^{-1/2}
  float* Y[2] = {ws + WS_Y0, ws + WS_Y1};
  float* Z[2] = {ws + WS_Z0, ws + WS_Z1};
  float* T    = ws + WS_T;
  for (int k = 0; k < NS_ITERS; ++k) {
    const int i = k & 1, o = i ^ 1;
    gemm512_kernel<0><<<G_DIM * 64, 128, 0, stream>>>(Z[i], Y[i], T);   // T  = 3I - Z*Y
    gemm512_kernel<1><<<G_DIM * 64, 128, 0, stream>>>(Y[i], T, Y[o]);   // Y' = 0.5*Y*T
    gemm512_kernel<1><<<G_DIM * 64, 128, 0, stream>>>(T, Z[i], Z[o]);   // Z' = 0.5*T*Z
  }
  // after an even number of iterations the result is back in Z0

  // 5. apply whitening + affine (WMMA f32 GEMM, N = 16384)
  apply_kernel<<<G_DIM * 8 * 256, 128, 0, stream>>>(x, weight, bias, ws, out);
}